// TargetOnlyAttention_86689619903514
// MI455X (gfx1250) — compile-verified
//
#include <hip/hip_runtime.h>
#include <hip/hip_bf16.h>

// Reference: v = x @ W.T + b ; out = broadcast(v[:,1,:], (B,T,C))
// => only token t=1 per batch feeds the Linear. Work = 8x1024x1024 GEMV + 64MB broadcast.
// MI455X: store-bandwidth bound (64MB / 23.3 TB/s ~ 2.9us). Use f32 WMMA for the GEMV.

#define B_DIM 8
#define T_DIM 2048
#define C_DIM 1024

typedef __attribute__((ext_vector_type(2))) float v2f;
typedef __attribute__((ext_vector_type(8))) float v8f;

// ---------------------------------------------------------------------------
// Kernel A: v1[m, n] = sum_k x[m, 1, k] * W[n, k] + bias[n]   (m = batch 0..7)
// One block per 16-wide N tile (64 blocks). 4 waves/block, each wave owns a
// K-quarter (256 k's = 64 x v_wmma_f32_16x16x4_f32), LDS cross-wave reduce.
// M is padded 8->16 with zero rows (lane-scale, no EXEC divergence: WMMA
// requires EXEC all-ones).
// ---------------------------------------------------------------------------
__global__ __launch_bounds__(128) void gemv_wmma_f32(
    const float* __restrict__ x,     // [B, T, C]
    const float* __restrict__ W,     // [C, C]  (row = output dim d, col = k)
    const float* __restrict__ bias,  // [C]
    float* __restrict__ v1)          // [B(=8), C] workspace
{
  const int lane = threadIdx.x & 31;
  const int wave = threadIdx.x >> 5;        // 0..3 : K-quarter
  const int n0   = blockIdx.x * 16;         // N tile base

  const int half = lane >> 4;               // 0: lanes 0-15, 1: lanes 16-31
  const int l15  = lane & 15;
  const int m    = l15;                     // A row for this lane (ISA layout)
  const int n    = n0 + l15;                // B column for this lane

  // A rows 8..15 are zero padding: load from row 0 but scale by 0.
  const float mscale = (m < B_DIM) ? 1.0f : 0.0f;
  const int   msafe  = (m < B_DIM) ? m : 0;
  const float* __restrict__ xrow = x + ((size_t)msafe * T_DIM + 1) * C_DIM;
  const float* __restrict__ wrow = W + (size_t)n * C_DIM;

  const int kBase = wave * (C_DIM / 4);     // 256 k's per wave

  v8f acc = {};
#pragma unroll 8
  for (int k = 0; k < C_DIM / 4; k += 4) {
    const int kk = kBase + k + 2 * half;    // half 0 -> K=0,1 ; half 1 -> K=2,3
    v2f a = *(const v2f*)(xrow + kk);       // A: 16x4 f32 (ISA 7.12.2 layout)
    a.x *= mscale;
    a.y *= mscale;
    v2f b = *(const v2f*)(wrow + kk);       // B: 4x16 f32, N striped over lanes
    // (neg_a, A, neg_b, B, c_mod, C, reuse_a, reuse_b)
    acc = __builtin_amdgcn_wmma_f32_16x16x4_f32(
        false, a, false, b, (short)0, acc, false, false);
  }

  // Cross-wave K reduction through LDS.
  __shared__ float red[4][32][8];
#pragma unroll
  for (int i = 0; i < 8; ++i) red[wave][lane][i] = acc[i];
  __syncthreads();

  if (wave == 0) {
    const float bn = bias[n];
#pragma unroll
    for (int i = 0; i < 8; ++i) {
      // D layout: VGPR i holds M = i (lanes 0-15) / M = i+8 (lanes 16-31)
      const int M = i + half * 8;
      if (M < B_DIM) {
        float s = red[0][lane][i] + red[1][lane][i] +
                  red[2][lane][i] + red[3][lane][i];
        v1[(size_t)M * C_DIM + n] = s + bn;
      }
    }
  }
}

// ---------------------------------------------------------------------------
// Kernel B: out[b, t, :] = v1[b, :] for all t. Pure wide-store bandwidth.
// Grid (T/16, B), 256 threads; each thread holds one float4 column chunk
// (256 * 4 = 1024 = C) and streams it to 16 token rows (b128 stores).
// ---------------------------------------------------------------------------
#define ROWS_PER_BLOCK 16

__global__ __launch_bounds__(256) void broadcast_rows(
    const float* __restrict__ v1,    // [8, C]
    float* __restrict__ out)         // [B, T, C]
{
  const int b    = blockIdx.y;
  const int t0   = blockIdx.x * ROWS_PER_BLOCK;
  const int col4 = threadIdx.x;      // 0..255

  const float4 val = ((const float4*)(v1 + (size_t)b * C_DIM))[col4];
  float4* dst = (float4*)(out + ((size_t)b * T_DIM + t0) * C_DIM) + col4;

#pragma unroll
  for (int r = 0; r < ROWS_PER_BLOCK; ++r) {
    dst[(size_t)r * (C_DIM / 4)] = val;
  }
}

extern "C" void kernel_launch(void* const* d_in, const int* in_sizes, int n_in,
                              void* d_out, int out_size, void* d_ws, size_t ws_size,
                              hipStream_t stream) {
  (void)in_sizes; (void)n_in; (void)out_size; (void)ws_size;

  const float* x    = (const float*)d_in[0];   // [8, 2048, 1024] f32
  const float* W    = (const float*)d_in[1];   // [1024, 1024] f32
  const float* bias = (const float*)d_in[2];   // [1024] f32
  float* out = (float*)d_out;                  // [8, 2048, 1024] f32
  float* v1  = (float*)d_ws;                   // [8, 1024] f32 scratch (32 KB)

  // GEMV via f32 WMMA: 64 N-tiles.
  gemv_wmma_f32<<<dim3(C_DIM / 16), dim3(128), 0, stream>>>(x, W, bias, v1);

  // Broadcast v1 across all T rows: (128, 8) blocks x 256 threads.
  broadcast_rows<<<dim3(T_DIM / ROWS_PER_BLOCK, B_DIM), dim3(256), 0, stream>>>(v1, out);
}